// MagPot_77283641524862
// MI455X (gfx1250) — compile-verified
//
#include <hip/hip_runtime.h>
#include <hip/hip_bf16.h>
#include <math.h>

#ifndef M_PI
#define M_PI 3.14159265358979323846
#endif

#define N_ATOMS    50000
#define N_EDGES    1600000
#define NMAX       8
#define BASIS      12
#define RCUT       6.0f
#define EMBED_DIM  16
#define N_STRUCT   64
#define ACC_STRIDE 64   // per-atom accumulator: [A:8][Bv:24][d1:8][Mv:24]

#define WAVES_PER_BLOCK 4
#define ATOMS_PER_WAVE  16

typedef __attribute__((ext_vector_type(2))) float v2f;
typedef __attribute__((ext_vector_type(8))) float v8f;

// ---------------------------------------------------------------------------
// Zero workspace accumulators and the 64 structure-energy outputs.
// ---------------------------------------------------------------------------
__global__ void magpot_zero(float* __restrict__ acc, float* __restrict__ out) {
  size_t i = (size_t)blockIdx.x * blockDim.x + threadIdx.x;
  const size_t tot = (size_t)N_ATOMS * ACC_STRIDE;
  if (i < tot) acc[i] = 0.0f;
  if (i < N_STRUCT) out[i] = 0.0f;
}

// ---------------------------------------------------------------------------
// Edge stage: Chebyshev radial basis * species-pair coeffs -> phi[8],
// scatter-add descriptors into per-atom accumulators (12.8MB, L2-resident).
// ---------------------------------------------------------------------------
__global__ __launch_bounds__(256) void magpot_edges(
    const float* __restrict__ pos, const float* __restrict__ mag,
    const float* __restrict__ shifts, const float* __restrict__ cheb,
    const int* __restrict__ species, const int* __restrict__ eidx,
    float* __restrict__ acc) {
  int e = blockIdx.x * blockDim.x + threadIdx.x;
  if (e >= N_EDGES) return;
  int i = eidx[e];
  int j = eidx[N_EDGES + e];

  float rx = pos[3*j+0] - pos[3*i+0] + shifts[3*e+0];
  float ry = pos[3*j+1] - pos[3*i+1] + shifts[3*e+1];
  float rz = pos[3*j+2] - pos[3*i+2] + shifts[3*e+2];
  float dist = sqrtf(rx*rx + ry*ry + rz*rz + 1e-12f);
  float inv = 1.0f / dist;
  float hx = rx*inv, hy = ry*inv, hz = rz*inv;

  float x = 2.0f * dist / RCUT - 1.0f;
  x = fminf(1.0f, fmaxf(-1.0f, x));
  float T[BASIS];
  T[0] = 1.0f; T[1] = x;
  #pragma unroll
  for (int b = 2; b < BASIS; ++b) T[b] = 2.0f*x*T[b-1] - T[b-2];
  float fc = (dist < RCUT) ? 0.5f * (cosf((float)M_PI * dist / RCUT) + 1.0f) : 0.0f;

  int pair = species[i] * 2 + species[j];
  const float* cb = cheb + pair * NMAX * BASIS;

  float mjx = mag[3*j+0], mjy = mag[3*j+1], mjz = mag[3*j+2];
  float mdr = mjx*hx + mjy*hy + mjz*hz;   // dot(m_j, rhat)

  float* a = acc + (size_t)i * ACC_STRIDE;
  #pragma unroll
  for (int n = 0; n < NMAX; ++n) {
    float p = 0.0f;
    #pragma unroll
    for (int b = 0; b < BASIS; ++b) p += T[b] * cb[n*BASIS + b];
    p *= fc;
    atomicAdd(a + n,            p);        // A
    atomicAdd(a + 8 + 3*n + 0,  p*hx);     // Bv
    atomicAdd(a + 8 + 3*n + 1,  p*hy);
    atomicAdd(a + 8 + 3*n + 2,  p*hz);
    atomicAdd(a + 32 + n,       p*mdr);    // d1
    atomicAdd(a + 40 + 3*n + 0, p*mjx);    // Mv (also yields d0 = m_i . Mv)
    atomicAdd(a + 40 + 3*n + 1, p*mjy);
    atomicAdd(a + 40 + 3*n + 2, p*mjz);
  }
}

// ---------------------------------------------------------------------------
// Wave-level GEMM on the f32 matrix pipe:  Out[16xNout] = act(X[16xK] @ W + b)
// Uses V_WMMA_F32_16X16X4_F32 (exact f32 math, matches reference precision).
// A layout: lane<16 -> row m=lane, K pair {k0,k0+1}; lane>=16 -> K {k0+2,k0+3}
// B layout: lane%16 -> col n, same K split.   C/D: m = r + 8*(lane>=16).
// ---------------------------------------------------------------------------
__device__ __forceinline__ float silu_f(float v) { return v / (1.0f + expf(-v)); }

__device__ __forceinline__ void wave_gemm_wmma(
    const float* __restrict__ Xs, int K, int ldx,
    const float* __restrict__ W, const float* __restrict__ bias, int Nout,
    float* __restrict__ Out, int ldo, bool act, int lane) {
  const int half = lane >> 4;
  const int l15  = lane & 15;
  for (int n0 = 0; n0 < Nout; n0 += 16) {
    v8f c;
    float bv = bias[n0 + l15];
    #pragma unroll
    for (int r = 0; r < 8; ++r) c[r] = bv;
    for (int k0 = 0; k0 < K; k0 += 4) {
      int ka = k0 + half * 2;
      v2f av, bvv;
      av.x  = Xs[l15 * ldx + ka];
      av.y  = Xs[l15 * ldx + ka + 1];
      bvv.x = W[(size_t)ka       * Nout + n0 + l15];
      bvv.y = W[(size_t)(ka + 1) * Nout + n0 + l15];
      c = __builtin_amdgcn_wmma_f32_16x16x4_f32(
          /*neg_a=*/false, av, /*neg_b=*/false, bvv,
          /*c_mod=*/(short)0, c, /*reuse_a=*/false, /*reuse_b=*/false);
    }
    #pragma unroll
    for (int r = 0; r < 8; ++r) {
      float v = c[r];
      if (act) v = silu_f(v);
      Out[(r + half * 8) * ldo + n0 + l15] = v;
    }
  }
}

// ---------------------------------------------------------------------------
// Atom stage: build descriptors, run both MLPs via WMMA, reduce per structure.
// One wave = 16 atoms; LDS double-buffered tiles per wave.
// ---------------------------------------------------------------------------
__global__ __launch_bounds__(WAVES_PER_BLOCK * 32) void magpot_atoms(
    const float* __restrict__ acc, const float* __restrict__ mag,
    const float* __restrict__ embed, const int* __restrict__ species,
    const int* __restrict__ batch,
    const float* __restrict__ sw0, const float* __restrict__ sb0,
    const float* __restrict__ sw1, const float* __restrict__ sb1,
    const float* __restrict__ sw2, const float* __restrict__ sb2,
    const float* __restrict__ mw0, const float* __restrict__ mb0,
    const float* __restrict__ mw1, const float* __restrict__ mb1,
    const float* __restrict__ mw2, const float* __restrict__ mb2,
    const float* __restrict__ eshift, float* __restrict__ out) {
  __shared__ float buf0[WAVES_PER_BLOCK][ATOMS_PER_WAVE * 64];
  __shared__ float buf1[WAVES_PER_BLOCK][ATOMS_PER_WAVE * 64];

  const int lane = threadIdx.x & 31;
  const int wid  = threadIdx.x >> 5;
  const int half = lane >> 4;
  const int l15  = lane & 15;
  const int base = (blockIdx.x * WAVES_PER_BLOCK + wid) * ATOMS_PER_WAVE;
  const int atom = base + l15;
  const bool valid = atom < N_ATOMS;
  const int  aq = valid ? atom : 0;            // clamped for safe loads
  const float* ar = acc + (size_t)aq * ACC_STRIDE;
  const int sp = species[aq];

  float* X = buf0[wid];
  float* H = buf1[wid];

  // ---- structural descriptors into X: [A(8) | |Bv|^2(8) | embed(16)], K=32
  if (half == 0) {
    #pragma unroll
    for (int n = 0; n < 8; ++n) X[l15*32 + n] = valid ? ar[n] : 0.0f;
    #pragma unroll
    for (int n = 0; n < 8; ++n) {
      float bx = ar[8+3*n+0], by = ar[8+3*n+1], bz = ar[8+3*n+2];
      X[l15*32 + 8 + n] = valid ? (bx*bx + by*by + bz*bz) : 0.0f;
    }
  } else {
    #pragma unroll
    for (int t = 0; t < 16; ++t)
      X[l15*32 + 16 + t] = valid ? embed[sp*EMBED_DIM + t] : 0.0f;
  }
  __syncthreads();
  wave_gemm_wmma(X, 32, 32, sw0, sb0, 64, H, 64, true, lane);   // 32 -> 64 silu
  __syncthreads();
  wave_gemm_wmma(H, 64, 64, sw1, sb1, 64, X, 64, true, lane);   // 64 -> 64 silu
  __syncthreads();
  float e_s = sb2[0];
  for (int k = 0; k < 64; ++k) e_s += X[l15*64 + k] * sw2[k];   // 64 -> 1 head

  // ---- magnetic descriptors into H: [d0(8) | d1(8) | |Mv|^2(8) | embed(16)], K=40
  float mix = mag[3*aq+0], miy = mag[3*aq+1], miz = mag[3*aq+2];
  if (half == 0) {
    #pragma unroll
    for (int n = 0; n < 8; ++n) {
      float mx = ar[40+3*n+0], my = ar[40+3*n+1], mz = ar[40+3*n+2];
      H[l15*40 + n]      = valid ? (mix*mx + miy*my + miz*mz) : 0.0f;  // d0
      H[l15*40 + 8 + n]  = valid ? ar[32 + n] : 0.0f;                  // d1
      H[l15*40 + 16 + n] = valid ? (mx*mx + my*my + mz*mz) : 0.0f;     // |Mv|^2
    }
  } else {
    #pragma unroll
    for (int t = 0; t < 16; ++t)
      H[l15*40 + 24 + t] = valid ? embed[sp*EMBED_DIM + t] : 0.0f;
  }
  __syncthreads();
  wave_gemm_wmma(H, 40, 40, mw0, mb0, 32, X, 32, true, lane);   // 40 -> 32 silu
  __syncthreads();
  wave_gemm_wmma(X, 32, 32, mw1, mb1, 32, H, 32, true, lane);   // 32 -> 32 silu
  __syncthreads();
  float e_m = mb2[0];
  for (int k = 0; k < 32; ++k) e_m += H[l15*32 + k] * mw2[k];   // 32 -> 1 head

  if (half == 0 && valid) {
    float e = e_s + e_m + eshift[sp];
    atomicAdd(out + batch[atom], e);
  }
}

// ---------------------------------------------------------------------------
// Host launch. Inputs in setup_inputs() dict order with params flattened in
// insertion order: cheb_coeff, embed, nn_struct[(w,b)x3], nn_mag[(w,b)x3],
// atomic_energy_shift.
// ---------------------------------------------------------------------------
extern "C" void kernel_launch(void* const* d_in, const int* in_sizes, int n_in,
                              void* d_out, int out_size, void* d_ws, size_t ws_size,
                              hipStream_t stream) {
  const float* pos    = (const float*)d_in[0];
  const float* mag    = (const float*)d_in[1];
  const float* shifts = (const float*)d_in[2];
  const float* cheb   = (const float*)d_in[3];
  const float* embed  = (const float*)d_in[4];
  const float* sw0 = (const float*)d_in[5];
  const float* sb0 = (const float*)d_in[6];
  const float* sw1 = (const float*)d_in[7];
  const float* sb1 = (const float*)d_in[8];
  const float* sw2 = (const float*)d_in[9];
  const float* sb2 = (const float*)d_in[10];
  const float* mw0 = (const float*)d_in[11];
  const float* mb0 = (const float*)d_in[12];
  const float* mw1 = (const float*)d_in[13];
  const float* mb1 = (const float*)d_in[14];
  const float* mw2 = (const float*)d_in[15];
  const float* mb2 = (const float*)d_in[16];
  const float* eshift = (const float*)d_in[17];
  const int* species  = (const int*)d_in[18];
  const int* eidx     = (const int*)d_in[19];
  const int* batch    = (const int*)d_in[20];

  float* out = (float*)d_out;
  float* acc = (float*)d_ws;   // N_ATOMS * 64 floats = 12.8 MB (L2-resident)

  {
    size_t tot = (size_t)N_ATOMS * ACC_STRIDE;
    int blocks = (int)((tot + 255) / 256);
    magpot_zero<<<blocks, 256, 0, stream>>>(acc, out);
  }
  magpot_edges<<<(N_EDGES + 255) / 256, 256, 0, stream>>>(
      pos, mag, shifts, cheb, species, eidx, acc);
  {
    int atoms_per_block = WAVES_PER_BLOCK * ATOMS_PER_WAVE;
    int blocks = (N_ATOMS + atoms_per_block - 1) / atoms_per_block;
    magpot_atoms<<<blocks, WAVES_PER_BLOCK * 32, 0, stream>>>(
        acc, mag, embed, species, batch,
        sw0, sb0, sw1, sb1, sw2, sb2,
        mw0, mb0, mw1, mb1, mw2, mb2,
        eshift, out);
  }
}